// MHSA3D_12369505812648
// MI455X (gfx1250) — compile-verified
//
#include <hip/hip_runtime.h>
#include <hip/hip_bf16.h>

// ---------------------------------------------------------------------------
// MHSA3D (transposed channel attention) for MI455X / gfx1250, fp32 end-to-end.
// GEMMs use V_WMMA_F32_16X16X4_F32; K5 stages V tiles through LDS with the
// Tensor Data Mover (tensor_load_to_lds + s_wait_tensorcnt) when available.
// ---------------------------------------------------------------------------

typedef __attribute__((ext_vector_type(2))) float v2f;
typedef __attribute__((ext_vector_type(8))) float v8f;
typedef __attribute__((ext_vector_type(4))) unsigned int v4u;
typedef __attribute__((ext_vector_type(8))) int v8i;
typedef __attribute__((ext_vector_type(4))) int v4i;

#define B_      4
#define T_      16
#define C_      31
#define H_      128
#define W_      128
#define HW_     (H_ * W_)                 // 16384
#define CHW_    (C_ * HW_)                // 507904  (stride between t-channels)
#define NB_     (T_ * CHW_)               // 8126464 (real per-batch elements)
#define NCOL_   (T_ * HW_)                // 262144  (flattened row length, 2^18)
#define ROWS_   C_                        // 31 real rows in [b,31,262144] view
#define PB_     (32 * NCOL_)              // 8388608 padded per-batch stride

__device__ __forceinline__ v8f wmma4(v2f a, v2f b, v8f c) {
  // D = A(16x4,f32) * B(4x16,f32) + C(16x16,f32)
  return __builtin_amdgcn_wmma_f32_16x16x4_f32(
      false, a, false, b, (short)0, c, false, false);
}

// ---------------------------------------------------------------------------
// K1: fused 1x1x1 QKV conv (16 -> 48 ch) + depthwise (1,3,3) conv.
// 16x16 output tile of one (b,d) plane; 18x18 halo of 48-ch qkv in LDS.
// q/k/v written in x layout [t][c][h][w] at padded per-batch stride PB_.
// ---------------------------------------------------------------------------
__global__ __launch_bounds__(256) void qkv_dw_kernel(
    const float* __restrict__ x, const float* __restrict__ wqkv,
    const float* __restrict__ wdw,
    float* __restrict__ qbuf, float* __restrict__ kbuf, float* __restrict__ vbuf) {
  __shared__ float lds[48 * 324];  // 62208 B
  const int b = blockIdx.z;
  const int d = blockIdx.y;
  const int th = blockIdx.x >> 3, tw = blockIdx.x & 7;
  const int tid = threadIdx.x;
  const long long xb = (long long)b * NB_ + (long long)d * HW_;  // + t*CHW_

  for (int p = tid; p < 324; p += 256) {
    const int hh = p / 18, ww = p % 18;
    const int h = th * 16 + hh - 1, w = tw * 16 + ww - 1;
    const bool inb = (h >= 0) & (h < H_) & (w >= 0) & (w < W_);
    const long long base = xb + (long long)h * W_ + w;
    float xv[16];
#pragma unroll
    for (int t = 0; t < 16; ++t)
      xv[t] = inb ? x[base + (long long)t * CHW_] : 0.0f;
#pragma unroll 4
    for (int o = 0; o < 48; ++o) {
      float a = 0.0f;
#pragma unroll
      for (int t = 0; t < 16; ++t) a = fmaf(wqkv[o * 16 + t], xv[t], a);
      lds[o * 324 + p] = a;
    }
  }
  __syncthreads();

  const int hh = tid >> 4, ww = tid & 15;
  const int h = th * 16 + hh, w = tw * 16 + ww;
  const long long obase = (long long)b * PB_ + (long long)d * HW_ +
                          (long long)h * W_ + w;  // + t*CHW_
#pragma unroll 4
  for (int o = 0; o < 48; ++o) {
    float s = 0.0f;
#pragma unroll
    for (int ky = 0; ky < 3; ++ky)
#pragma unroll
      for (int kx = 0; kx < 3; ++kx)
        s = fmaf(wdw[o * 9 + ky * 3 + kx],
                 lds[o * 324 + (hh + ky) * 18 + (ww + kx)], s);
    float* dst = (o < 16) ? qbuf : ((o < 32) ? kbuf : vbuf);
    dst[obase + (long long)(o & 15) * CHW_] = s;
  }
}

// ---------------------------------------------------------------------------
// K2: per-row sum of squares for q (rows 0..123) and k (rows 124..247).
// ---------------------------------------------------------------------------
__global__ __launch_bounds__(256) void sumsq_kernel(
    const float* __restrict__ qbuf, const float* __restrict__ kbuf,
    float* __restrict__ sums) {
  const int row = blockIdx.x;    // 0..247
  const int chunk = blockIdx.y;  // 0..7
  const float* base;
  if (row < 124)
    base = qbuf + (long long)(row / 31) * PB_ + (long long)(row % 31) * NCOL_;
  else {
    const int r = row - 124;
    base = kbuf + (long long)(r / 31) * PB_ + (long long)(r % 31) * NCOL_;
  }
  base += (long long)chunk * 32768;
  float s = 0.0f;
  for (int i = threadIdx.x; i < 32768; i += 256) {
    const float v = base[i];
    s = fmaf(v, v, s);
  }
  __shared__ float red[256];
  red[threadIdx.x] = s;
  __syncthreads();
  for (int k = 128; k > 0; k >>= 1) {
    if (threadIdx.x < k) red[threadIdx.x] += red[threadIdx.x + k];
    __syncthreads();
  }
  if (threadIdx.x == 0) atomicAdd(&sums[row], red[0]);
}

// ---------------------------------------------------------------------------
// K3: raw logits S[b] = Q[32 x 262144] * K^T (rows 31 are zero padding),
// f32 WMMA 16x16x4, K split across 512 waves/batch, atomic reduction.
// All loads unconditional thanks to the padded row -> tight wmma loop.
// ---------------------------------------------------------------------------
__global__ __launch_bounds__(256) void attn_logits_kernel(
    const float* __restrict__ qbuf, const float* __restrict__ kbuf,
    float* __restrict__ logits) {
  const int b = blockIdx.y;
  const int wave = threadIdx.x >> 5;
  const int lane = threadIdx.x & 31;
  const int chunk = blockIdx.x * 8 + wave;  // 0..511
  const long long kstart = (long long)chunk * 512;
  const int m = lane & 15;
  const int kh = lane >> 4;
  const float* q0 = qbuf + (long long)b * PB_ + (long long)m * NCOL_ + kstart + 2 * kh;
  const float* q1 = q0 + (long long)16 * NCOL_;
  const float* k0 = kbuf + (long long)b * PB_ + (long long)m * NCOL_ + kstart + 2 * kh;
  const float* k1 = k0 + (long long)16 * NCOL_;

  v8f acc00 = {}, acc01 = {}, acc10 = {}, acc11 = {};
#pragma unroll 2
  for (int kk = 0; kk < 512; kk += 4) {
    const v2f a0 = *(const v2f*)(q0 + kk);
    const v2f a1 = *(const v2f*)(q1 + kk);
    const v2f b0 = *(const v2f*)(k0 + kk);
    const v2f b1 = *(const v2f*)(k1 + kk);
    acc00 = wmma4(a0, b0, acc00);
    acc01 = wmma4(a0, b1, acc01);
    acc10 = wmma4(a1, b0, acc10);
    acc11 = wmma4(a1, b1, acc11);
  }
  float* lg = logits + b * 1024;
  const int N = lane & 15;
  const int Mb = 8 * (lane >> 4);
#pragma unroll
  for (int v = 0; v < 8; ++v) {
    const int M = Mb + v;
    atomicAdd(lg + M * 32 + N, acc00[v]);
    atomicAdd(lg + M * 32 + 16 + N, acc01[v]);
    atomicAdd(lg + (16 + M) * 32 + N, acc10[v]);
    atomicAdd(lg + (16 + M) * 32 + 16 + N, acc11[v]);
  }
}

// ---------------------------------------------------------------------------
// K4: fold 1/(||q||*||k||)*temperature into logits, softmax over 31, pad.
// ---------------------------------------------------------------------------
__global__ void softmax_kernel(const float* __restrict__ sums,
                               const float* __restrict__ temperature,
                               const float* __restrict__ logits,
                               float* __restrict__ attn) {
  const int t = threadIdx.x;  // one block of 128
  if (t < 124) {
    const int b = t / 31, c = t % 31;
    const float temp = temperature[0];
    const float nq = fmaxf(sqrtf(sums[b * 31 + c]), 1e-12f);
    float vals[31];
    float mx = -3.0e38f;
    for (int d = 0; d < 31; ++d) {
      const float nk = fmaxf(sqrtf(sums[124 + b * 31 + d]), 1e-12f);
      const float v = logits[b * 1024 + c * 32 + d] / (nq * nk) * temp;
      vals[d] = v;
      mx = fmaxf(mx, v);
    }
    float sum = 0.0f;
    for (int d = 0; d < 31; ++d) {
      const float e = expf(vals[d] - mx);
      vals[d] = e;
      sum += e;
    }
    const float inv = 1.0f / sum;
    for (int d = 0; d < 31; ++d) attn[b * 1024 + c * 32 + d] = vals[d] * inv;
    attn[b * 1024 + c * 32 + 31] = 0.0f;
  } else if (t < 128) {
    const int b = t - 124;
    for (int d = 0; d < 32; ++d) attn[b * 1024 + 31 * 32 + d] = 0.0f;
  }
}

// ---------------------------------------------------------------------------
// K5: AO[b] = attn[32x32] * V[32 x 262144].  Each block TDM-loads a 32x128
// f32 tile of V into LDS (16 KB), then 8 waves run WMMA from LDS fragments.
// ---------------------------------------------------------------------------
__global__ __launch_bounds__(256) void attn_apply_kernel(
    const float* __restrict__ attn, const float* __restrict__ vbuf,
    float* __restrict__ ao) {
  __shared__ float vt[32 * 128];  // 16 KB tile: rows 0..31, 128 columns
  const int b = blockIdx.y;
  const int wave = threadIdx.x >> 5;
  const int lane = threadIdx.x & 31;
  const long long col0 = (long long)blockIdx.x * 128;  // block's column base
  const float* vb = vbuf + (long long)b * PB_;

#if __has_builtin(__builtin_amdgcn_tensor_load_to_lds)
  if (wave == 0) {
    // Tensor DMA descriptor (D#): 2D tensor, 4-byte elems,
    // tile = 128 x 32 rows, row stride = NCOL_.
    const unsigned long long ga =
        (unsigned long long)(const void*)(vb + col0);
    const unsigned ldsa = (unsigned)(unsigned long long)(const void*)vt;
    v4u g0;
    g0.x = 1u;                                   // count=1, user mode
    g0.y = ldsa;                                 // lds_addr
    g0.z = (unsigned)(ga & 0xffffffffull);       // global_addr[31:0]
    g0.w = (unsigned)((ga >> 32) & 0x1ffffffull) // global_addr[56:32]
           | (2u << 30);                         // type=2 ("image")
    v8i g1;
    g1.s0 = 0x00020000;            // data_size=2 (4B), no pad/iter/multicast
    g1.s1 = (int)((NCOL_ & 0xffffu) << 16);          // tensor_dim0 lo16 @63:48
    g1.s2 = (int)(((32u & 0xffffu) << 16) |          // tensor_dim1 lo16
                  ((NCOL_ >> 16) & 0xffffu));        // tensor_dim0 hi16
    g1.s3 = (int)(128u << 16);     // tile_dim0=128 @127:112 (dim1 hi16 = 0)
    g1.s4 = 32;                    // tile_dim1=32 (tile_dim2=0)
    g1.s5 = (int)(NCOL_ & 0xffffffffu);              // dim0_stride[31:0]
    g1.s6 = 0;                                        // dim0_stride hi / d1 lo
    g1.s7 = 0;
    v4i gz = {0, 0, 0, 0};
#if defined(__clang_major__) && (__clang_major__ >= 23)
    v8i gz8 = {0, 0, 0, 0, 0, 0, 0, 0};
    __builtin_amdgcn_tensor_load_to_lds(g0, g1, gz, gz, gz8, 0);
#else
    __builtin_amdgcn_tensor_load_to_lds(g0, g1, gz, gz, 0);
#endif
    __builtin_amdgcn_s_wait_tensorcnt(0);
  }
  __syncthreads();
#else
  // Fallback: cooperative vector loads into LDS.
  for (int i = threadIdx.x; i < 32 * 128 / 4; i += 256) {
    const int r = i / 32, c4 = (i % 32) * 4;
    *(float4*)&vt[r * 128 + c4] = *(const float4*)(vb + (long long)r * NCOL_ + col0 + c4);
  }
  __syncthreads();
#endif

  const int m = lane & 15;
  const int kh = lane >> 4;
  const float* at = attn + b * 1024;
  v2f afr0[8], afr1[8];
#pragma unroll
  for (int s = 0; s < 8; ++s) {
    const int kk = 4 * s + 2 * kh;
    afr0[s] = (v2f){at[m * 32 + kk], at[m * 32 + kk + 1]};
    afr1[s] = (v2f){at[(16 + m) * 32 + kk], at[(16 + m) * 32 + kk + 1]};
  }
  const int nloc = wave * 16 + m;  // column within the 128-wide LDS tile
  v8f acc0 = {}, acc1 = {};
#pragma unroll
  for (int s = 0; s < 8; ++s) {
    const int r0 = 4 * s + 2 * kh;  // <= 30; r0+1 <= 31 (zero pad row)
    const v2f bf = (v2f){vt[r0 * 128 + nloc], vt[(r0 + 1) * 128 + nloc]};
    acc0 = wmma4(afr0[s], bf, acc0);
    acc1 = wmma4(afr1[s], bf, acc1);
  }
  float* ob = ao + (long long)b * PB_;
  const long long n = col0 + wave * 16 + (lane & 15);
  const int Mb = 8 * (lane >> 4);
#pragma unroll
  for (int v = 0; v < 8; ++v) {
    ob[(long long)(Mb + v) * NCOL_ + n] = acc0[v];
    ob[(long long)(16 + Mb + v) * NCOL_ + n] = acc1[v];  // row 31 -> pad row
  }
}

// ---------------------------------------------------------------------------
// K6: out = w_proj(16x16) * AO viewed as [b][16][507904].
// ---------------------------------------------------------------------------
__global__ __launch_bounds__(256) void proj_kernel(
    const float* __restrict__ ao, const float* __restrict__ wproj,
    float* __restrict__ out) {
  const int b = blockIdx.y;
  const int wave = threadIdx.x >> 5;
  const int lane = threadIdx.x & 31;
  const int tile = blockIdx.x * 8 + wave;  // 0..31743
  const long long s0 = (long long)tile * 16;
  const float* ab = ao + (long long)b * PB_;
  float* ob = out + (long long)b * NB_;
  const int m = lane & 15;
  const int kh = lane >> 4;

  v2f afr[4];
#pragma unroll
  for (int s = 0; s < 4; ++s) {
    const int kk = 4 * s + 2 * kh;
    afr[s] = (v2f){wproj[m * 16 + kk], wproj[m * 16 + kk + 1]};
  }
  v8f acc = {};
#pragma unroll
  for (int s = 0; s < 4; ++s) {
    const int i0 = 4 * s + 2 * kh;
    const v2f bf = (v2f){ab[(long long)i0 * CHW_ + s0 + m],
                         ab[(long long)(i0 + 1) * CHW_ + s0 + m]};
    acc = wmma4(afr[s], bf, acc);
  }
  const int N = lane & 15;
  const int Mb = 8 * (lane >> 4);
#pragma unroll
  for (int v = 0; v < 8; ++v)
    ob[(long long)(Mb + v) * CHW_ + s0 + N] = acc[v];
}

// ---------------------------------------------------------------------------
extern "C" void kernel_launch(void* const* d_in, const int* in_sizes, int n_in,
                              void* d_out, int out_size, void* d_ws,
                              size_t ws_size, hipStream_t stream) {
  const float* x = (const float*)d_in[0];       // [4,16,31,128,128]
  const float* wqkv = (const float*)d_in[1];    // [48,16]
  const float* wdw = (const float*)d_in[2];     // [48,1,1,3,3]
  const float* wproj = (const float*)d_in[3];   // [16,16]
  const float* temp = (const float*)d_in[4];    // [1]
  float* out = (float*)d_out;

  float* ws = (float*)d_ws;
  float* qbuf = ws;                       // 4*PB_ floats (reused as AO)
  float* kbuf = qbuf + (size_t)B_ * PB_;
  float* vbuf = kbuf + (size_t)B_ * PB_;
  float* sums = vbuf + (size_t)B_ * PB_;  // 256 floats (248 used)
  float* logits = sums + 256;             // 4*32*32
  float* attn = logits + 4096;            // 4*32*32

  hipMemsetAsync(sums, 0, (256 + 4096) * sizeof(float), stream);
  // Zero the padded row 31 of each batch for q/k/v (enables guard-free WMMA).
  for (int b = 0; b < B_; ++b) {
    const size_t off = (size_t)b * PB_ + (size_t)ROWS_ * NCOL_;
    hipMemsetAsync(qbuf + off, 0, NCOL_ * sizeof(float), stream);
    hipMemsetAsync(kbuf + off, 0, NCOL_ * sizeof(float), stream);
    hipMemsetAsync(vbuf + off, 0, NCOL_ * sizeof(float), stream);
  }

  qkv_dw_kernel<<<dim3(64, 31, 4), 256, 0, stream>>>(x, wqkv, wdw, qbuf, kbuf,
                                                     vbuf);
  sumsq_kernel<<<dim3(248, 8), 256, 0, stream>>>(qbuf, kbuf, sums);
  attn_logits_kernel<<<dim3(64, 4), 256, 0, stream>>>(qbuf, kbuf, logits);
  softmax_kernel<<<1, 128, 0, stream>>>(sums, temp, logits, attn);
  attn_apply_kernel<<<dim3(2048, 4), 256, 0, stream>>>(attn, vbuf, qbuf);
  proj_kernel<<<dim3(3968, 4), 256, 0, stream>>>(qbuf, wproj, out);
}